// xLSTM_63763084476830
// MI455X (gfx1250) — compile-verified
//
#include <hip/hip_runtime.h>

// Problem constants (from reference): B=16, T=512, I=512, H=16, E=32
#define B_  16
#define T_  512
#define I_  512
#define H_  16
#define E_  32

typedef __attribute__((ext_vector_type(2))) float v2f;
typedef __attribute__((ext_vector_type(4))) float v4f;
typedef __attribute__((ext_vector_type(8))) float v8f;

// ---------------------------------------------------------------------------
// CDNA5 async copy: per-lane 16B global -> LDS, tracked on ASYNCcnt
// (CDNA5 ISA). LDS byte offset = low 32 bits of the
// generic pointer (LDS aperture: addr[31:0] = LDS offset).
// ---------------------------------------------------------------------------
__device__ __forceinline__ void async_ld16(const float* lds_dst, const float* g_src) {
  unsigned           loff = (unsigned)(size_t)lds_dst;
  unsigned long long ga   = (unsigned long long)(size_t)g_src;
  asm volatile("global_load_async_to_lds_b128 %0, %1, off"
               :: "v"(loff), "v"(ga) : "memory");
}
__device__ __forceinline__ void wait_async0() {
  asm volatile("s_wait_asynccnt 0x0" ::: "memory");
}

// ---------------------------------------------------------------------------
// Kernel 1: mg[bt,l,e] = sum_i x[bt,i]*W_ml[l,i,e] + b_ml[l,e]
// Blocked WMMA GEMM. Workgroup (512 thr = 16 waves) computes a macro-tile:
//   M_WG=128 rows x L_WG=8 heads x E=32 cols (128 16x16 tiles).
// Wave = 2 msubs x 2 l x 2 ntiles = 8 tiles (64 acc VGPRs):
//   A reuse x4, B reuse x2 -> 6 ds_load_b64 per 8 WMMAs, zero operand movs.
// LDS layouts (bank-conflict-free by construction, 64 banks x 4B):
//   xs : [r][k] with row stride 36 dwords (pad 4) -> A b64 loads hit 64
//        distinct banks (36r mod 64 distinct for r=0..15; +2-dword upper
//        half never collides since 36*dr mod 64 is a multiple of 4).
//   wsp: k-pair interleaved [l][kp][e][2]; odd-kp rows rotated by 32 dwords
//        so the two lane halves (which always read opposite-parity kp) use
//        disjoint bank sets 0-31 / 32-63.
// x chunk async-staged (ASYNCcnt path); W chunk staged synchronously because
// of the interleave. Traffic: x read 8x (128MB), W read 64x (256MB).
// Per-lane WMMA f32 16x16x4 layout (ISA 7.12.2):
//   row/col = lane&15, khalf = lane>>4
//   A: a = {A[row][2*khalf], A[row][2*khalf+1]}
//   B: b = {B[2*khalf][col], B[2*khalf+1][col]}
//   D: c[r] = D[r + 8*khalf][col]
// ---------------------------------------------------------------------------
#define KC        32
#define M_WG      128
#define L_WG      8
#define XS_STRIDE 36                       // KC + 4 dword pad

__global__ __launch_bounds__(512) void mg_gemm_wmma(const float* __restrict__ x,
                                                    const float* __restrict__ Wml,
                                                    const float* __restrict__ bml,
                                                    float* __restrict__ mg) {
  __shared__ float xs[M_WG * XS_STRIDE];        // 18 KB
  __shared__ float wsp[L_WG * (KC / 2) * 64];   // 32 KB  [l][kp][64 dwords]

  const int tid   = threadIdx.x;
  const int lane  = tid & 31;
  const int wave  = tid >> 5;              // 0..15

  const int mgrp  = blockIdx.x >> 3;       // 0..63   (8192 / 128)
  const int lgrp  = blockIdx.x & 7;        // 0..7    (64 / 8)
  const int m0    = mgrp * M_WG;
  const int lbase = lgrp * L_WG;

  const int mp    = wave >> 2;             // 0..3 -> rows [mp*32, +32)
  const int lp    = wave & 3;              // 0..3 -> l pair {2lp, 2lp+1}

  const int row   = lane & 15;
  const int khalf = lane >> 4;
  const int colc  = lane & 15;

  v8f acc[8];                              // [mi*4 + l2*2 + nt]
  #pragma unroll
  for (int ti = 0; ti < 8; ++ti) acc[ti] = (v8f){};

  for (int kc = 0; kc < I_; kc += KC) {
    // ---- stage x chunk (async): 1024 x 16B segments, 2 per thread ----
    #pragma unroll
    for (int i = 0; i < 2; ++i) {
      const int s  = tid + i * 512;
      const int r  = s >> 3;               // 0..127
      const int sk = (s & 7) * 4;          // 0,4,..,28
      async_ld16(&xs[r * XS_STRIDE + sk], x + (size_t)(m0 + r) * I_ + kc + sk);
    }
    // ---- stage W chunk (sync, k-pair interleave + rotation) ----
    // 1024 tasks: task = (l, kp, eseg); 2 per thread. Each task reads the
    // 4-float e-segments of rows k=2kp and k=2kp+1 and interleaves them.
    #pragma unroll
    for (int i = 0; i < 2; ++i) {
      const int s   = tid + i * 512;
      const int l   = s >> 7;              // 0..7
      const int rem = s & 127;
      const int kp  = rem >> 3;            // 0..15
      const int e   = (rem & 7) * 4;       // 0,4,..,28
      const float* gb = Wml + (size_t)(lbase + l) * I_ * E_
                            + (size_t)(kc + 2 * kp) * E_ + e;
      const v4f g0 = *(const v4f*)(gb);        // k = 2kp
      const v4f g1 = *(const v4f*)(gb + E_);   // k = 2kp+1
      float* dst = &wsp[((l * (KC / 2) + kp) << 6) + (e << 1) + ((kp & 1) << 5)];
      #pragma unroll
      for (int j = 0; j < 4; ++j) {
        dst[2 * j]     = g0[j];
        dst[2 * j + 1] = g1[j];
      }
    }
    wait_async0();
    __syncthreads();

    // ---- compute: 8 K-steps x 8 tiles ----
    for (int ks = 0; ks < KC; ks += 4) {
      const int k0 = ks + 2 * khalf;
      const int kp = k0 >> 1;              // lane halves have opposite parity
      v2f a0 = *(const v2f*)(&xs[(mp * 32 + row) * XS_STRIDE + k0]);
      v2f a1 = *(const v2f*)(&xs[(mp * 32 + 16 + row) * XS_STRIDE + k0]);
      #pragma unroll
      for (int l2 = 0; l2 < 2; ++l2) {
        const int lrow = ((lp * 2 + l2) * (KC / 2) + kp) << 6;
        #pragma unroll
        for (int nt = 0; nt < 2; ++nt) {
          const int col = colc + nt * 16;
          v2f bb = *(const v2f*)(&wsp[lrow + (((col << 1) + ((kp & 1) << 5)) & 63)]);
          acc[0 * 4 + l2 * 2 + nt] = __builtin_amdgcn_wmma_f32_16x16x4_f32(
              false, a0, false, bb, (short)0, acc[0 * 4 + l2 * 2 + nt], false, false);
          acc[1 * 4 + l2 * 2 + nt] = __builtin_amdgcn_wmma_f32_16x16x4_f32(
              false, a1, false, bb, (short)0, acc[1 * 4 + l2 * 2 + nt], false, false);
        }
      }
    }
    __syncthreads();
  }

  // ---- epilogue: bias + store. mg row stride over bt = 4H*E = 2048 ----
  #pragma unroll
  for (int mi = 0; mi < 2; ++mi) {
    #pragma unroll
    for (int l2 = 0; l2 < 2; ++l2) {
      #pragma unroll
      for (int nt = 0; nt < 2; ++nt) {
        const int lg     = lbase + lp * 2 + l2;
        const int col    = colc + nt * 16;
        const float bias = bml[lg * E_ + col];
        float* out = mg + (size_t)(m0 + mp * 32 + mi * 16) * (4 * H_ * E_)
                        + (size_t)lg * E_ + col;
        const v8f c = acc[mi * 4 + l2 * 2 + nt];
        #pragma unroll
        for (int r = 0; r < 8; ++r) {
          const int m = r + 8 * khalf;
          out[(size_t)m * (4 * H_ * E_)] = c[r] + bias;
        }
      }
    }
  }
}

// ---------------------------------------------------------------------------
// Kernel 2: lg[bt,g] = sum_i x[bt,i] * W_lin[g,i] + b_lin[g]
// M=8192, N=32, K=512; W_lin is transposed so the B operand is a contiguous
// float2 along i. Tiny (~32MB traffic) -- one wave per 16x16 tile.
// ---------------------------------------------------------------------------
__global__ __launch_bounds__(256) void lg_gemm_wmma(const float* __restrict__ x,
                                                    const float* __restrict__ Wlin,
                                                    const float* __restrict__ blin,
                                                    float* __restrict__ lg) {
  const int lane  = threadIdx.x & 31;
  const int wave  = threadIdx.x >> 5;
  const int tile  = blockIdx.x * 8 + wave;       // 0 .. 1023
  const int ntile = tile & 1;
  const int mtile = tile >> 1;

  const int row   = lane & 15;
  const int khalf = lane >> 4;
  const int col   = (lane & 15) + ntile * 16;    // g index 0..31

  const float* xrow = x    + (size_t)(mtile * 16 + row) * I_;
  const float* wg   = Wlin + (size_t)col * I_;

  v8f c = {};
  #pragma unroll 4
  for (int kb = 0; kb < I_; kb += 4) {
    const int k0 = kb + 2 * khalf;
    v2f a = *(const v2f*)(xrow + k0);
    v2f b = *(const v2f*)(wg + k0);
    c = __builtin_amdgcn_wmma_f32_16x16x4_f32(false, a, false, b,
                                              (short)0, c, false, false);
  }

  const float bias = blin[col];
  float* out = lg + (size_t)(mtile * 16) * (2 * H_) + col;
  #pragma unroll
  for (int r = 0; r < 8; ++r) {
    const int m = r + 8 * khalf;
    out[(size_t)m * (2 * H_)] = c[r] + bias;
  }
}

// ---------------------------------------------------------------------------
// Kernel 3: sequential mLSTM scan. One wave per (b,h); lane j owns C[:,j]
// (32 VGPRs) and n[j]. h_t is output-only, so num/den/sigmoid run only at
// t = T-1. d_out: [0, B*H*E) = h_last; then C_f (B,H,E,E).
// ---------------------------------------------------------------------------
__global__ __launch_bounds__(256) void mlstm_scan(const float* __restrict__ mg,
                                                  const float* __restrict__ lg,
                                                  float* __restrict__ out) {
  const int lane = threadIdx.x & 31;             // j / E index
  const int wave = threadIdx.x >> 5;
  const int bh   = blockIdx.x * 8 + wave;        // 0 .. 255
  const int b    = bh >> 4;
  const int h    = bh & 15;

  float m = 0.0f;
  float n = 0.0f;
  float C[E_];
  #pragma unroll
  for (int i = 0; i < E_; ++i) C[i] = 0.0f;

  float hout = 0.0f;

  for (int t = 0; t < T_; ++t) {
    const float* g    = mg + (size_t)(b * T_ + t) * (4 * H_) * E_;
    const float  vv   = g[(H_     + h) * E_ + lane];
    const float  kk   = g[(2 * H_ + h) * E_ + lane];
    const float* lgbt = lg + (size_t)(b * T_ + t) * (2 * H_);
    const float  it   = lgbt[h];
    const float  ft   = lgbt[H_ + h];

    const float m_new = fmaxf(ft + m, it);
    const float i_g   = __expf(it - m_new);
    const float f_g   = __expf(ft + m - m_new);
    m = m_new;

    n = f_g * n + i_g * kk;
    const float igk = i_g * kk;

    if (t == T_ - 1) {
      const float oo = g[(0      + h) * E_ + lane];
      const float qq = g[(3 * H_ + h) * E_ + lane];
      float num = 0.0f;
      #pragma unroll
      for (int i = 0; i < E_; ++i) {
        const float vi = __shfl(vv, i, 32);
        const float qi = __shfl(qq, i, 32);
        C[i] = f_g * C[i] + igk * vi;
        num += C[i] * qi;
      }
      float d = n * qq;
      #pragma unroll
      for (int off = 16; off > 0; off >>= 1)
        d += __shfl_xor(d, off, 32);
      d = fmaxf(d, 1.0f);
      const float sig = 1.0f / (1.0f + __expf(-oo));
      hout = sig * num / d;
    } else {
      #pragma unroll
      for (int i = 0; i < E_; ++i) {
        const float vi = __shfl(vv, i, 32);
        C[i] = f_g * C[i] + igk * vi;
      }
    }
  }

  out[(size_t)bh * E_ + lane] = hout;
  float* Cf = out + (size_t)B_ * H_ * E_ + (size_t)bh * E_ * E_;
  #pragma unroll
  for (int i = 0; i < E_; ++i)
    Cf[(size_t)i * E_ + lane] = C[i];
}

// ---------------------------------------------------------------------------
extern "C" void kernel_launch(void* const* d_in, const int* in_sizes, int n_in,
                              void* d_out, int out_size, void* d_ws, size_t ws_size,
                              hipStream_t stream) {
  const float* x    = (const float*)d_in[0];   // (B,T,I)
  const float* Wml  = (const float*)d_in[1];   // (4H,I,E)
  const float* bml  = (const float*)d_in[2];   // (4H,E)
  const float* Wlin = (const float*)d_in[3];   // (2H,I)
  const float* blin = (const float*)d_in[4];   // (2H,)

  float* out = (float*)d_out;
  // Workspace: mg (B*T*4H*E = 16,777,216 floats = 64 MB) then lg (1 MB).
  float* mg = (float*)d_ws;
  float* lg = mg + (size_t)B_ * T_ * 4 * H_ * E_;

  // 64 m-groups x 8 l-groups, 512 threads (16 waves) each
  mg_gemm_wmma<<<512, 512, 0, stream>>>(x, Wml, bml, mg);
  // 1024 tile-waves
  lg_gemm_wmma<<<128, 256, 0, stream>>>(x, Wlin, blin, lg);
  // 256 (b,h) recurrences, 8 waves per block
  mlstm_scan<<<32, 256, 0, stream>>>(mg, lg, out);
}